// EmissionAbsorptionRaymarcherStratified_45878840656476
// MI455X (gfx1250) — compile-verified
//
#include <hip/hip_runtime.h>

// EmissionAbsorption raymarcher for MI455X (gfx1250).
// Memory-bound: ~293 MB moved @ 23.3 TB/s => ~12.6 us floor.
//   - TDM (tensor_load_to_lds, th:NT) streams each ray's 16KB feature tile
//     into LDS, double-buffered, overlapped with the wave32 product scan.
//   - Einsum is a per-ray matvec (no shared-B GEMM => WMMA inapplicable);
//     fp32 FMAs from LDS are exact and need only ~23 TFLOPS.
//   - All global traffic is touch-once => non-temporal hints everywhere.

typedef unsigned int u32;
typedef unsigned long long u64;
typedef u32 v4u __attribute__((ext_vector_type(4)));
typedef u32 v8u __attribute__((ext_vector_type(8)));
typedef float v2f __attribute__((ext_vector_type(2)));

#define WAVES_PER_BLOCK 4
#define NPTS 64
#define NFEAT 64
#define TILE_PTS 32
#define TILE_ELEMS (TILE_PTS * NFEAT) // 2048 floats = 8 KB

__global__ __launch_bounds__(WAVES_PER_BLOCK * 32)
void ea_raymarch_tdm_kernel(const float* __restrict__ dens,
                            const float* __restrict__ feat,
                            float* __restrict__ out0,   // (nrays, 65)
                            float* __restrict__ outw,   // (nrays, 64)
                            int nrays)
{
    // 4 waves * 2 buffers * 8KB = 64 KB feature staging + 1KB weights
    __shared__ __align__(16) float sfeat[WAVES_PER_BLOCK][2][TILE_ELEMS];
    __shared__ __align__(16) float sw[WAVES_PER_BLOCK][NPTS];

    const int tid  = (int)threadIdx.x;
    const int wave = tid >> 5;
    const int lane = tid & 31;
    const int ray  = (int)blockIdx.x * WAVES_PER_BLOCK + wave;
    if (ray >= nrays) return;

    // ---------------- TDM: kick off both feature-tile DMAs first ----------
    // D# group0: [1:0]=count=1, [63:32]=lds_addr, [120:64]=global_addr,
    //            [127:126]=type=2.
    // D# group1: data_size=2 (4B), tensor_dim0=64, tensor_dim1=32,
    //            tile_dim0=64, tile_dim1=32, tensor_dim0_stride=64.
    const u64 fbase = (u64)(const void*)feat + (u64)ray * (NPTS * NFEAT * 4);
#pragma unroll
    for (int t = 0; t < 2; ++t) {
        const u64 ga  = fbase + (u64)t * (TILE_PTS * NFEAT * 4);
        const u32 lds = (u32)(size_t)&sfeat[wave][t][0]; // low32 of flat = LDS byte addr
        v4u g0;
        g0.x = 1u;                               // count=1, user descriptor
        g0.y = lds;                              // lds_addr
        g0.z = (u32)ga;                          // global_addr[31:0]
        g0.w = (u32)(ga >> 32) | (2u << 30);     // global_addr[56:32] | type=2
        v8u g1;
        g1[0] = (2u << 16);                      // data_size=4B; wg_mask=0
        g1[1] = ((u32)NFEAT) << 16;              // tensor_dim0[15:0]=64
        g1[2] = ((u32)TILE_PTS) << 16;           // tensor_dim0 hi=0 | tensor_dim1 lo=32
        g1[3] = ((u32)NFEAT) << 16;              // tensor_dim1 hi=0 | tile_dim0=64
        g1[4] = (u32)TILE_PTS;                   // tile_dim1=32 | tile_dim2=0
        g1[5] = (u32)NFEAT;                      // tensor_dim0_stride=64 (elements)
        g1[6] = 0u;                              // stride0 hi | stride1 lo
        g1[7] = 0u;                              // stride1 hi
        asm volatile("tensor_load_to_lds %0, %1 th:TH_LOAD_NT"
                     :: "s"(g0), "s"(g1) : "memory");
    }

    // ---------------- weights: exclusive product scan (overlaps DMA) ------
    // lane l holds d[2l], d[2l+1]
    const v2f dd =
        __builtin_nontemporal_load((const v2f*)&dens[(u64)ray * NPTS + 2 * lane]);
    const float d0 = dd.x, d1 = dd.y;
    const float t0 = 1.0f + 1e-10f - d0;
    const float t1 = 1.0f + 1e-10f - d1;

    float incl = t0 * t1;                        // pair product
#pragma unroll
    for (int off = 1; off < 32; off <<= 1) {
        float up = __shfl_up(incl, off, 32);
        if (lane >= off) incl *= up;
    }
    float excl = __shfl_up(incl, 1, 32);         // exclusive pair-scan
    if (lane == 0) excl = 1.0f;

    const float w0 = d0 * excl;                  // weights[2l]
    const float w1 = d1 * excl * t0;             // weights[2l+1]

    // opacity = 1 - prod(1-d)
    float q = (1.0f - d0) * (1.0f - d1);
#pragma unroll
    for (int off = 1; off < 32; off <<= 1)
        q *= __shfl_xor(q, off, 32);
    const float opacity = 1.0f - q;

    // store weights: global (coalesced 8B/lane, NT) + LDS copy for broadcast
    v2f wp; wp.x = w0; wp.y = w1;
    __builtin_nontemporal_store(wp, (v2f*)&outw[(u64)ray * NPTS + 2 * lane]);
    *(v2f*)&sw[wave][2 * lane] = wp;

    // ---------------- einsum: features[f] = sum_p w[p] * F[p,f] -----------
    // Half-wave g handles the contiguous point range [32t + 16g, 32t + 16g + 15],
    // so each lane's 16 weights are a contiguous 64B run -> 4x ds_load_b128.
    const int g  = lane >> 4;
    const int qd = lane & 15;
    float ax = 0.f, ay = 0.f, az = 0.f, aw = 0.f;

#pragma unroll
    for (int t = 0; t < 2; ++t) {
        if (t == 0) asm volatile("s_wait_tensorcnt 1" ::: "memory");
        else        asm volatile("s_wait_tensorcnt 0" ::: "memory");
        const float4* fb  = (const float4*)&sfeat[wave][t][0];
        const float4* wv4 = (const float4*)&sw[wave][32 * t + 16 * g];
        float4 wreg[4];
        wreg[0] = wv4[0]; wreg[1] = wv4[1]; wreg[2] = wv4[2]; wreg[3] = wv4[3];
        const float* wf = (const float*)wreg;
#pragma unroll
        for (int i = 0; i < 16; ++i) {
            // p = 32*t + 16*g + i
            const float  wv = wf[i];                              // register (const idx)
            const float4 f  = fb[(16 * g + i) * (NFEAT / 4) + qd]; // ds_load_b128
            ax = fmaf(wv, f.x, ax);
            ay = fmaf(wv, f.y, ay);
            az = fmaf(wv, f.z, az);
            aw = fmaf(wv, f.w, aw);
        }
    }

    // combine halves: lane l (<16) += lane l+16
    // (group 0 summed points {0..15, 32..47}, group 1 {16..31, 48..63})
    ax += __shfl_down(ax, 16, 32);
    ay += __shfl_down(ay, 16, 32);
    az += __shfl_down(az, 16, 32);
    aw += __shfl_down(aw, 16, 32);

    const u64 rb = (u64)ray * (NFEAT + 1);
    if (lane < 16) {
        __builtin_nontemporal_store(ax, &out0[rb + 4 * lane + 0]);
        __builtin_nontemporal_store(ay, &out0[rb + 4 * lane + 1]);
        __builtin_nontemporal_store(az, &out0[rb + 4 * lane + 2]);
        __builtin_nontemporal_store(aw, &out0[rb + 4 * lane + 3]);
    }
    if (lane == 0) __builtin_nontemporal_store(opacity, &out0[rb + NFEAT]);
}

extern "C" void kernel_launch(void* const* d_in, const int* in_sizes, int n_in,
                              void* d_out, int out_size, void* d_ws, size_t ws_size,
                              hipStream_t stream) {
    const float* dens = (const float*)d_in[0];   // (2,16384,64,1) fp32
    const float* feat = (const float*)d_in[1];   // (2,16384,64,64) fp32
    float* out0 = (float*)d_out;                 // (nrays,65) then (nrays,64)
    const int nrays = in_sizes[0] / NPTS;        // 32768
    float* outw = out0 + (size_t)nrays * (NFEAT + 1);

    const int blocks = (nrays + WAVES_PER_BLOCK - 1) / WAVES_PER_BLOCK;
    ea_raymarch_tdm_kernel<<<dim3(blocks), dim3(WAVES_PER_BLOCK * 32), 0, stream>>>(
        dens, feat, out0, outw, nrays);
}